// CGBN_cuda_3813930959613
// MI455X (gfx1250) — compile-verified
//
#include <hip/hip_runtime.h>
#include <math.h>

typedef float v2f __attribute__((ext_vector_type(2)));
typedef float v8f __attribute__((ext_vector_type(8)));

// ---------------- compile-time tables (triples in reference order) -----------
constexpr int cTRI_L [22] = {0,0,0,0, 1,1,1,1,1, 2,2,2,2,2,2,2, 3,3,3,3,3,3};
constexpr int cTRI_L1[22] = {0,1,2,3, 1,1,2,3,3, 1,2,2,2,3,3,3, 2,2,3,3,3,3};
constexpr int cTRI_L2[22] = {0,1,2,3, 0,1,1,2,3, 1,0,1,2,1,2,3, 1,2,0,1,2,3};
constexpr int cTRI_CG[22] = {0,1,10,35, 84,93,120,165,270,
                             417,462,487,562,687,792,967,
                             1212,1317,1492,1541,1688,1933};
constexpr int cLEV[5] = {0,4,9,16,22};
constexpr int cST [4] = {0,16,64,144};          // activation / output row offset per level
constexpr int cWST[4] = {0,16384,36864,65536};  // W row offset per level
constexpr int cML [4] = {1024,1280,1792,1536};  // middle taus per level
constexpr int cSST[4] = {0,1024,2304,4096};     // channel offset per level

// runtime-indexed copies (for the tiny CG/W-scale kernels)
__constant__ int gTRI_L [22] = {0,0,0,0, 1,1,1,1,1, 2,2,2,2,2,2,2, 3,3,3,3,3,3};
__constant__ int gTRI_L1[22] = {0,1,2,3, 1,1,2,3,3, 1,2,2,2,3,3,3, 2,2,3,3,3,3};
__constant__ int gTRI_L2[22] = {0,1,2,3, 0,1,1,2,3, 1,0,1,2,1,2,3, 1,2,0,1,2,3};
__constant__ int gTRI_CG[22] = {0,1,10,35, 84,93,120,165,270,
                                417,462,487,562,687,792,967,
                                1212,1317,1492,1541,1688,1933};
__constant__ int gWST[4] = {0,16384,36864,65536};
__constant__ int gML [4] = {1024,1280,1792,1536};
__constant__ int gSST[4] = {0,1024,2304,4096};

#define N_CHAN  5632
#define N_WROWS 90112
#define BATCH   2048

// workspace float offsets
#define WS_CG    0
#define WS_SSQ   2304
#define WS_INV   7936
#define WS_WSC   13568   // float2-indexed from here (even -> 8B aligned)

__device__ __forceinline__ v8f wmma4(v2f a, v2f b, v8f c) {
    return __builtin_amdgcn_wmma_f32_16x16x4_f32(false, a, false, b, (short)0, c, false, false);
}

// ---------------- CG coefficients (device, double precision) -----------------
__device__ double dfact(int n) { double r = 1.0; for (int i = 2; i <= n; ++i) r *= (double)i; return r; }

__device__ float cg_coef(int l1, int m1, int l2, int m2, int l, int m) {
    if (m1 + m2 != m) return 0.0f;
    double pre = sqrt((double)(2 * l + 1) * dfact(l1 + l2 - l) * dfact(l1 - l2 + l) *
                      dfact(-l1 + l2 + l) / dfact(l1 + l2 + l + 1));
    pre *= sqrt(dfact(l + m) * dfact(l - m) * dfact(l1 - m1) * dfact(l1 + m1) *
                dfact(l2 - m2) * dfact(l2 + m2));
    double s = 0.0;
    for (int k = 0; k <= l1 + l2 - l; ++k) {
        int a0 = k, a1 = l1 + l2 - l - k, a2 = l1 - m1 - k, a3 = l2 + m2 - k;
        int a4 = l - l2 + m1 + k, a5 = l - l1 - m2 + k;
        if (a0 < 0 || a1 < 0 || a2 < 0 || a3 < 0 || a4 < 0 || a5 < 0) continue;
        double den = dfact(a0) * dfact(a1) * dfact(a2) * dfact(a3) * dfact(a4) * dfact(a5);
        s += ((k & 1) ? -1.0 : 1.0) / den;
    }
    return (float)(pre * s);
}

__global__ void k_cg(float* cg) {
    int i = threadIdx.x;
    if (i >= 22) return;
    int l = gTRI_L[i], l1 = gTRI_L1[i], l2 = gTRI_L2[i];
    int d = 2 * l + 1, d1 = 2 * l1 + 1, d2 = 2 * l2 + 1;
    float* dst = cg + gTRI_CG[i];
    for (int a = 0; a < d1; ++a)
        for (int b = 0; b < d2; ++b)
            for (int p = 0; p < d; ++p)
                dst[(a * d2 + b) * d + p] = cg_coef(l1, a - l1, l2, b - l2, l, p - l);
}

__global__ void k_init(float* sumsq) {
    int i = blockIdx.x * blockDim.x + threadIdx.x;
    if (i < N_CHAN) sumsq[i] = 0.0f;
}

// ---------------- pass 1: per-channel sum of |mid|^2 (fully unrolled) --------
template<int TR>
__device__ __forceinline__ void stats_triple(const float2* __restrict__ actf2,
                                             const float* __restrict__ cg,
                                             float* __restrict__ sumsq,
                                             int tid, int t, int s, int bg) {
    constexpr int l = cTRI_L[TR], l1 = cTRI_L1[TR], l2 = cTRI_L2[TR];
    constexpr int d = 2 * l + 1, d1 = 2 * l1 + 1, d2 = 2 * l2 + 1;
    constexpr int cgoff = cTRI_CG[TR];
    float acc = 0.0f;
    #pragma unroll 1
    for (int bb = 0; bb < 8; ++bb) {
        int b = bg * 8 + bb;
        float2 F1[d1], F2[d2];
        int base1 = b * 256 + cST[l1] + t * d1;
        int base2 = b * 256 + cST[l2] + s * d2;
        #pragma unroll
        for (int m = 0; m < d1; ++m) F1[m] = actf2[base1 + m];
        #pragma unroll
        for (int m = 0; m < d2; ++m) F2[m] = actf2[base2 + m];
        float mr[d] = {}, mi[d] = {};
        #pragma unroll
        for (int a = 0; a < d1; ++a) {
            #pragma unroll
            for (int c2 = 0; c2 < d2; ++c2) {
                int p = a + c2 - l1 - l2 + l;       // folds at compile time
                if (p >= 0 && p < d) {
                    float cgv = cg[cgoff + (a * d2 + c2) * d + p];
                    float pr = F1[a].x * F2[c2].x - F1[a].y * F2[c2].y;
                    float pi = F1[a].x * F2[c2].y + F1[a].y * F2[c2].x;
                    mr[p] += cgv * pr;
                    mi[p] += cgv * pi;
                }
            }
        }
        #pragma unroll
        for (int p = 0; p < d; ++p) acc += mr[p] * mr[p] + mi[p] * mi[p];
    }
    atomicAdd(&sumsq[TR * 256 + tid], acc);
}

template<int TR, int END>
struct SChain {
    static __device__ __forceinline__ void run(const float2* a, const float* cg, float* sq,
                                               int tid, int t, int s, int bg) {
        stats_triple<TR>(a, cg, sq, tid, t, s, bg);
        SChain<TR + 1, END>::run(a, cg, sq, tid, t, s, bg);
    }
};
template<int END>
struct SChain<END, END> {
    static __device__ __forceinline__ void run(const float2*, const float*, float*, int, int, int, int) {}
};

__global__ __launch_bounds__(256) void k_stats(const float* __restrict__ act,
                                               const float* __restrict__ cg,
                                               float* __restrict__ sumsq) {
    const float2* actf2 = reinterpret_cast<const float2*>(act);
    int tid = threadIdx.x;
    int t = tid >> 4, s = tid & 15;
    int bg = blockIdx.x;                 // 8 batches per block
    SChain<0, 22>::run(actf2, cg, sumsq, tid, t, s, bg);
}

// ---------------- pass 2: inv_std, W scaled by inv_std -----------------------
__global__ void k_finalize(const float* __restrict__ sumsq,
                           const float* __restrict__ bn,
                           float* __restrict__ inv) {
    int i = blockIdx.x * blockDim.x + threadIdx.x;
    if (i >= N_CHAN) return;
    int l = (i < 1024) ? 0 : (i < 2304) ? 1 : (i < 4096) ? 2 : 3;
    float d = (float)(2 * l + 1);
    float bstd = sqrtf(sumsq[i] / (2048.0f * d));
    float nstd = 0.5f * (bn[i] + bstd);  // BM_CNT = 1.0
    inv[i] = 1.0f / (nstd + 1e-5f);
}

__global__ void k_wscale(const float* __restrict__ W,
                         const float* __restrict__ inv,
                         float* __restrict__ wsc) {
    int i = blockIdx.x * blockDim.x + threadIdx.x;
    if (i >= N_WROWS) return;
    int l = (i < 16384) ? 0 : (i < 36864) ? 1 : (i < 65536) ? 2 : 3;
    int r = i - gWST[l];
    int c = r % gML[l];
    float iv = inv[gSST[l] + c];
    const float2* wf2 = reinterpret_cast<const float2*>(W);
    float2* of2 = reinterpret_cast<float2*>(wsc);
    float2 w = wf2[i];
    of2[i] = make_float2(w.x * iv, w.y * iv);
}

// ---------------- pass 3: fused (Wsc (x) F1) via WMMA, then (x) F2 + CG ------
template<int L, int TR>
__device__ __forceinline__ void out_triple(const float2* __restrict__ actf2,
                                           const float2* __restrict__ wscf2,
                                           const float* __restrict__ cg,
                                           float* __restrict__ U,
                                           float* __restrict__ outacc,
                                           int tid, int wave, int lane, int b0) {
    constexpr int nb = 2;
    constexpr int l1 = cTRI_L1[TR], l2 = cTRI_L2[TR];
    constexpr int d1 = 2 * l1 + 1, d2 = 2 * l2 + 1, dl = 2 * L + 1;
    constexpr int st1 = cST[l1], st2 = cST[l2];
    constexpr int wst = cWST[L], Ml = cML[L];
    constexpr int localc = TR * 256 - cSST[L];
    constexpr int cgoff = cTRI_CG[TR];

    int colL = lane & 15;
    int khalf = (lane >> 4) * 2;
    int o = wave;

    // ---- GEMM1: U[o,s,(b,m1)] = sum_t Wsc[o,t,s] (*c) F1[b,t,m1] ----
    int bl_c = colL / d1;                  // division by constant
    int m1_c = colL - bl_c * d1;
    bool colok = (bl_c < nb);
    int bcol = b0 + (colok ? bl_c : 0);    // clamped -> always-valid address

    v8f cre = {}; v8f cim = {};
    #pragma unroll
    for (int kk = 0; kk < 16; kk += 4) {
        int t0 = kk + khalf;
        int abase = wst + o * Ml + localc + (t0 << 4) + colL;   // A[M=s, K=t]
        float2 wa0 = wscf2[abase];
        float2 wa1 = wscf2[abase + 16];
        int bbase = bcol * 256 + st1 + t0 * d1 + m1_c;          // B[K=t, N=col]
        float2 fb0 = actf2[bbase];
        float2 fb1 = actf2[bbase + d1];
        v2f are = {wa0.x, wa1.x};
        v2f aim = {wa0.y, wa1.y};
        v2f ain = {-wa0.y, -wa1.y};
        v2f bre = {colok ? fb0.x : 0.0f, colok ? fb1.x : 0.0f}; // cndmask, no EXEC branch
        v2f bim = {colok ? fb0.y : 0.0f, colok ? fb1.y : 0.0f};
        cre = wmma4(are, bre, cre);
        cre = wmma4(ain, bim, cre);
        cim = wmma4(are, bim, cim);
        cim = wmma4(aim, bre, cim);
    }
    // D layout: lane -> N = lane&15, VGPR r -> M = r + (lane>>4)*8
    #pragma unroll
    for (int r = 0; r < 8; ++r) {
        int srw = r + ((lane >> 4) << 3);
        int uidx = ((o * 16 + srw) * 16 + colL) * 2;
        U[uidx]     = cre[r];
        U[uidx + 1] = cim[r];
    }
    __syncthreads();

    // ---- stage 2: G[m1,n] = sum_s U[o,s,(b,m1)] (*c) F2[b,s,n]; CG fold ----
    constexpr int cnt = nb * 16 * d2;
    if (tid < cnt) {
        int n2 = tid % d2;                 // constant divisors
        int o2 = (tid / d2) & 15;
        int bl = tid / (16 * d2);
        int b = b0 + bl;
        float2 f2v[16];
        int base2 = b * 256 + st2 + n2;
        #pragma unroll
        for (int s2 = 0; s2 < 16; ++s2) f2v[s2] = actf2[base2 + s2 * d2];
        float gr[d1] = {}, gi[d1] = {};
        #pragma unroll
        for (int s2 = 0; s2 < 16; ++s2) {
            float2 fv = f2v[s2];
            int ubase = ((o2 * 16 + s2) * 16 + bl * d1) * 2;
            #pragma unroll
            for (int m1 = 0; m1 < d1; ++m1) {
                float ur = U[ubase + m1 * 2];
                float ui = U[ubase + m1 * 2 + 1];
                gr[m1] += ur * fv.x - ui * fv.y;
                gi[m1] += ur * fv.y + ui * fv.x;
            }
        }
        #pragma unroll
        for (int m1 = 0; m1 < d1; ++m1) {
            int p = m1 + n2 - l1 - l2 + L;
            if (p >= 0 && p < dl) {
                float cgv = cg[cgoff + (m1 * d2 + n2) * dl + p];
                int oidx = ((bl * 16 + o2) * dl + p) * 2;
                atomicAdd(&outacc[oidx],     cgv * gr[m1]);
                atomicAdd(&outacc[oidx + 1], cgv * gi[m1]);
            }
        }
    }
    __syncthreads();
}

template<int L, int TR, int END>
struct OChain {
    static __device__ __forceinline__ void run(const float2* a, const float2* w, const float* cg,
                                               float* U, float* oa, int tid, int wave, int lane, int b0) {
        out_triple<L, TR>(a, w, cg, U, oa, tid, wave, lane, b0);
        OChain<L, TR + 1, END>::run(a, w, cg, U, oa, tid, wave, lane, b0);
    }
};
template<int L, int END>
struct OChain<L, END, END> {
    static __device__ __forceinline__ void run(const float2*, const float2*, const float*,
                                               float*, float*, int, int, int, int) {}
};

// grid.x = 1024 batch groups (2 batches each), 512 thr (16 waves, wave = o)
template<int L>
__global__ __launch_bounds__(512) void k_out_t(const float* __restrict__ act,
                                               const float* __restrict__ wsc,
                                               const float* __restrict__ cg,
                                               float* __restrict__ out) {
    constexpr int nb = 2, dl = 2 * L + 1;
    const float2* actf2 = reinterpret_cast<const float2*>(act);
    const float2* wscf2 = reinterpret_cast<const float2*>(wsc);
    float2* outf2 = reinterpret_cast<float2*>(out);

    int tid = threadIdx.x;
    int wave = tid >> 5;
    int lane = tid & 31;
    int b0 = blockIdx.x * nb;

    __shared__ float U[16 * 16 * 16 * 2];          // [o][s][col] complex (32 KB)
    __shared__ float outacc[nb * 16 * dl * 2];

    if (tid < nb * 16 * dl * 2) outacc[tid] = 0.0f;
    __syncthreads();

    OChain<L, cLEV[L], cLEV[L + 1]>::run(actf2, wscf2, cg, U, outacc, tid, wave, lane, b0);

    if (tid < nb * 16 * dl) {
        int bl = tid / (16 * dl);
        int rem = tid - bl * (16 * dl);            // = o*dl + p
        int row = (b0 + bl) * 256 + cST[L] + rem;
        outf2[row] = make_float2(outacc[tid * 2], outacc[tid * 2 + 1]);
    }
}

// ---------------- launcher ----------------------------------------------------
extern "C" void kernel_launch(void* const* d_in, const int* in_sizes, int n_in,
                              void* d_out, int out_size, void* d_ws, size_t ws_size,
                              hipStream_t stream) {
    const float* act = (const float*)d_in[0];   // (2048, 256, 2)
    const float* W   = (const float*)d_in[1];   // (90112, 2)
    const float* bn  = (const float*)d_in[2];   // (5632,)
    float* out = (float*)d_out;
    float* ws  = (float*)d_ws;

    float* cg    = ws + WS_CG;
    float* sumsq = ws + WS_SSQ;
    float* inv   = ws + WS_INV;
    float* wsc   = ws + WS_WSC;

    k_init    <<<(N_CHAN + 255) / 256, 256, 0, stream>>>(sumsq);
    k_cg      <<<1, 32, 0, stream>>>(cg);
    k_stats   <<<BATCH / 8, 256, 0, stream>>>(act, cg, sumsq);
    k_finalize<<<(N_CHAN + 255) / 256, 256, 0, stream>>>(sumsq, bn, inv);
    k_wscale  <<<(N_WROWS + 255) / 256, 256, 0, stream>>>(W, inv, wsc);

    k_out_t<0><<<BATCH / 2, 512, 0, stream>>>(act, wsc, cg, out);
    k_out_t<1><<<BATCH / 2, 512, 0, stream>>>(act, wsc, cg, out);
    k_out_t<2><<<BATCH / 2, 512, 0, stream>>>(act, wsc, cg, out);
    k_out_t<3><<<BATCH / 2, 512, 0, stream>>>(act, wsc, cg, out);
}